// DanClassifier_48198122995720
// MI455X (gfx1250) — compile-verified
//
#include <hip/hip_runtime.h>
#include <hip/hip_bf16.h>

typedef __attribute__((ext_vector_type(2))) float v2f;
typedef __attribute__((ext_vector_type(8))) float v8f;

#define VOCAB   100000
#define D_IN    300
#define B_DOCS  256
#define L_TOK   1000
#define H_DIM   512
#define C_OUT   5

#define SPLIT   8
#define TOK_PER (L_TOK / SPLIT)   // 125 tokens per block
#define GBLK    96                // 3 waves; lanes 0..74 active (float4 columns)

#define AVG_STRIDE 300            // 300 % 64 = 44, gcd(44,64)=4 -> 16 distinct banks for m=0..15
#define H_STRIDE   516            // 516 % 64 = 4 -> perfect 64-bank spread for b64 A reads

// ---------------------------------------------------------------------------
// Kernel 1: fused gather + partial segment-sum.
// grid = (B_DOCS, SPLIT), block = 96. Lane q < 75 owns float4 column q of the
// 300-float embedding row. Each token row is read as 75 contiguous 16B loads.
// ---------------------------------------------------------------------------
__global__ __launch_bounds__(GBLK) void dan_gather_sum(
    const float* __restrict__ emb, const int* __restrict__ docs,
    float* __restrict__ partial)
{
  const int b = blockIdx.x;
  const int chunk = blockIdx.y;
  __shared__ int s_idx[TOK_PER];
  for (int i = threadIdx.x; i < TOK_PER; i += GBLK)
    s_idx[i] = docs[b * L_TOK + chunk * TOK_PER + i];
  __syncthreads();

  const int q = threadIdx.x;         // float4 column index
  if (q < D_IN / 4) {
    float4 acc = make_float4(0.f, 0.f, 0.f, 0.f);
    #pragma unroll 4
    for (int i = 0; i < TOK_PER; ++i) {
      if (i + 8 < TOK_PER)           // speculative prefetch ~8 tokens ahead
        __builtin_prefetch(emb + (size_t)s_idx[i + 8] * D_IN + 4 * q, 0, 1);
      const float4 v =
          *(const float4*)(emb + (size_t)s_idx[i] * D_IN + 4 * q);
      acc.x += v.x; acc.y += v.y; acc.z += v.z; acc.w += v.w;
    }
    float4* dst =
        (float4*)(partial + ((size_t)b * SPLIT + chunk) * D_IN + 4 * q);
    *dst = acc;
  }
}

// ---------------------------------------------------------------------------
// Kernel 2: reduce partials -> avg, then 3-layer MLP via V_WMMA_F32_16X16X4_F32.
// grid = 16 (row tiles of 16), block = 256 (8 waves). Each wave owns 4 of the
// 32 N-tiles of the 512-wide hidden layers. Activations staged in dynamic LDS.
// ---------------------------------------------------------------------------
__global__ __launch_bounds__(256) void dan_mlp_wmma(
    const float* __restrict__ partial, const int* __restrict__ doc_lens,
    const float* __restrict__ W1, const float* __restrict__ b1,
    const float* __restrict__ W2, const float* __restrict__ b2,
    const float* __restrict__ W3, const float* __restrict__ b3,
    float* __restrict__ out)
{
  extern __shared__ float smem[];
  float* lds_avg = smem;                        // 16 * 300
  float* lds_h1  = smem + 16 * AVG_STRIDE;      // 16 * 516
  float* lds_h2  = lds_h1 + 16 * H_STRIDE;      // 16 * 516

  const int rt   = blockIdx.x;                  // row tile (0..15)
  const int tid  = threadIdx.x;
  const int w    = tid >> 5;                    // wave 0..7
  const int lane = tid & 31;
  const int nl   = lane & 15;                   // N (B/C/D) or M (A) within tile
  const int kq   = lane >> 4;                   // K-pair selector

  // ---- Phase 0: reduce SPLIT partials, divide by doc_len -> lds_avg -------
  for (int t = tid; t < 16 * D_IN; t += 256) {
    const int m = t / D_IN, d = t - m * D_IN;
    const int row = rt * 16 + m;
    float s = 0.f;
    #pragma unroll
    for (int c = 0; c < SPLIT; ++c)
      s += partial[((size_t)row * SPLIT + c) * D_IN + d];
    lds_avg[m * AVG_STRIDE + d] = s / (float)doc_lens[row];
  }
  __syncthreads();

  // ---- Phase 1: h1 = relu(avg @ W1 + b1); K = 300 = 75 * 4 ----------------
  {
    v8f acc[4] = {};
    for (int kk = 0; kk < D_IN / 4; ++kk) {
      const int k0 = 4 * kk + 2 * kq;
      v2f a;
      a.x = lds_avg[nl * AVG_STRIDE + k0];
      a.y = lds_avg[nl * AVG_STRIDE + k0 + 1];
      #pragma unroll
      for (int i = 0; i < 4; ++i) {
        const int n = (w + 8 * i) * 16 + nl;
        v2f bb;
        bb.x = W1[(size_t)k0 * H_DIM + n];
        bb.y = W1[(size_t)(k0 + 1) * H_DIM + n];
        acc[i] = __builtin_amdgcn_wmma_f32_16x16x4_f32(
            false, a, false, bb, (short)0, acc[i], false, false);
      }
    }
    #pragma unroll
    for (int i = 0; i < 4; ++i) {
      const int n = (w + 8 * i) * 16 + nl;
      const float bias = b1[n];
      #pragma unroll
      for (int r = 0; r < 8; ++r) {
        const int mm = r + 8 * kq;
        lds_h1[mm * H_STRIDE + n] = fmaxf(acc[i][r] + bias, 0.f);
      }
    }
  }
  __syncthreads();

  // ---- Phase 2: h2 = relu(h1 @ W2 + b2); K = 512 = 128 * 4 ----------------
  {
    v8f acc[4] = {};
    for (int kk = 0; kk < H_DIM / 4; ++kk) {
      const int k0 = 4 * kk + 2 * kq;
      v2f a;
      a.x = lds_h1[nl * H_STRIDE + k0];
      a.y = lds_h1[nl * H_STRIDE + k0 + 1];
      #pragma unroll
      for (int i = 0; i < 4; ++i) {
        const int n = (w + 8 * i) * 16 + nl;
        v2f bb;
        bb.x = W2[(size_t)k0 * H_DIM + n];
        bb.y = W2[(size_t)(k0 + 1) * H_DIM + n];
        acc[i] = __builtin_amdgcn_wmma_f32_16x16x4_f32(
            false, a, false, bb, (short)0, acc[i], false, false);
      }
    }
    #pragma unroll
    for (int i = 0; i < 4; ++i) {
      const int n = (w + 8 * i) * 16 + nl;
      const float bias = b2[n];
      #pragma unroll
      for (int r = 0; r < 8; ++r) {
        const int mm = r + 8 * kq;
        lds_h2[mm * H_STRIDE + n] = fmaxf(acc[i][r] + bias, 0.f);
      }
    }
  }
  __syncthreads();

  // ---- Phase 3: out = h2 @ W3 + b3; one masked 16-wide tile (C_OUT = 5) ---
  if (w == 0) {
    v8f acc = {};
    for (int kk = 0; kk < H_DIM / 4; ++kk) {
      const int k0 = 4 * kk + 2 * kq;
      v2f a;
      a.x = lds_h2[nl * H_STRIDE + k0];
      a.y = lds_h2[nl * H_STRIDE + k0 + 1];
      v2f bb;
      bb.x = (nl < C_OUT) ? W3[(size_t)k0 * C_OUT + nl] : 0.f;
      bb.y = (nl < C_OUT) ? W3[(size_t)(k0 + 1) * C_OUT + nl] : 0.f;
      acc = __builtin_amdgcn_wmma_f32_16x16x4_f32(
          false, a, false, bb, (short)0, acc, false, false);
    }
    if (nl < C_OUT) {
      const float bias = b3[nl];
      #pragma unroll
      for (int r = 0; r < 8; ++r) {
        const int mm = r + 8 * kq;
        out[(size_t)(rt * 16 + mm) * C_OUT + nl] = acc[r] + bias;
      }
    }
  }
}

// ---------------------------------------------------------------------------
extern "C" void kernel_launch(void* const* d_in, const int* in_sizes, int n_in,
                              void* d_out, int out_size, void* d_ws, size_t ws_size,
                              hipStream_t stream) {
  const float* emb = (const float*)d_in[0];
  const float* W1  = (const float*)d_in[1];
  const float* b1  = (const float*)d_in[2];
  const float* W2  = (const float*)d_in[3];
  const float* b2  = (const float*)d_in[4];
  const float* W3  = (const float*)d_in[5];
  const float* b3  = (const float*)d_in[6];
  const int* docs  = (const int*)d_in[7];
  const int* dlen  = (const int*)d_in[8];
  float* out       = (float*)d_out;
  float* partial   = (float*)d_ws;     // B_DOCS * SPLIT * D_IN floats = 2.4 MB

  dim3 g1(B_DOCS, SPLIT);
  dan_gather_sum<<<g1, GBLK, 0, stream>>>(emb, docs, partial);

  const size_t shmem =
      (size_t)(16 * AVG_STRIDE + 2 * 16 * H_STRIDE) * sizeof(float);
  dan_mlp_wmma<<<16, 256, shmem, stream>>>(partial, dlen, W1, b1, W2, b2,
                                           W3, b3, out);
}